// DotLayer_16647293239323
// MI455X (gfx1250) — compile-verified
//
#include <hip/hip_runtime.h>
#include <hip/hip_bf16.h>
#include <math.h>

// DotLayer: out[b,h] = tanh( sum_{d<128} X[b,d]*W[b,d,h] + W[b,128,h] )
// W[b] = AUX[b,:] viewed as [129][128] row-major.
//
// Memory-bound (549 MB @ 23.3 TB/s ~= 24 us floor; only 0.27 GFLOP).
// Strategy: one wave32 per sample, stream AUX exactly once, fully coalesced,
// and run the matvec on the matrix pipe via V_WMMA_F32_16X16X4_F32 with the
// Y vector replicated across all 16 M rows of the A operand.

typedef __attribute__((ext_vector_type(2))) float v2f;
typedef __attribute__((ext_vector_type(8))) float v8f;

#define D_DIM 128
#define H_DIM 128
#define AUX_STRIDE ((D_DIM + 1) * H_DIM)   // 16512
#define WAVES_PER_BLOCK 8

__global__ __launch_bounds__(32 * WAVES_PER_BLOCK)
void DotLayer_wmma_kernel(const float* __restrict__ X,
                          const float* __restrict__ AUX,
                          float* __restrict__ out,
                          int Btot) {
    const int lane = threadIdx.x & 31;
    const int wave = threadIdx.x >> 5;
    const int b    = blockIdx.x * WAVES_PER_BLOCK + wave;
    if (b >= Btot) return;               // wave-uniform: EXEC all-ones inside

    const int half = lane >> 4;          // 0 -> K pair (0,1); 1 -> K pair (2,3)
    const int n    = lane & 15;          // column within 16-wide h tile

    const float* __restrict__ Xb = X   + (size_t)b * D_DIM;
    const float* __restrict__ Wb = AUX + (size_t)b * AUX_STRIDE;  // [129][128]

    // A operand (16x4 f32, all 16 M rows = Y chunk):
    // lane needs Y[4i + 2*half], Y[4i + 2*half + 1]  -> one contiguous b64.
    const float* __restrict__ Apt = Xb + 2 * half;

    for (int t = 0; t < 4; ++t) {
        const int h0 = 32 * t;           // two 16-wide tiles per iteration
        // B operand (4x16 f32): lane(n) -> W[k0+2*half, h0+n], W[k0+2*half+1, h0+n]
        const float* __restrict__ Bpt0 = Wb + (size_t)(2 * half) * H_DIM + h0 + n;
        const float* __restrict__ Bpt1 = Bpt0 + 16;

        v8f c0 = {};
        v8f c1 = {};
#pragma unroll 4
        for (int i = 0; i < 32; ++i) {   // K chunks of 4 over d = 0..127
            v2f a = *(const v2f*)(Apt + 4 * i);          // 8B aligned
            v2f b0, b1;
            b0.x = Bpt0[i * 4 * H_DIM];
            b0.y = Bpt0[i * 4 * H_DIM + H_DIM];
            b1.x = Bpt1[i * 4 * H_DIM];
            b1.y = Bpt1[i * 4 * H_DIM + H_DIM];
            c0 = __builtin_amdgcn_wmma_f32_16x16x4_f32(
                     false, a, false, b0, (short)0, c0, false, false);
            c1 = __builtin_amdgcn_wmma_f32_16x16x4_f32(
                     false, a, false, b1, (short)0, c1, false, false);
        }
        // D rows are identical (A rows replicated). c[0] on lane L is element
        // (M = 8*half, N = n) -> value for h tile of that accumulator.
        // half 0 emits tile 2t, half 1 emits tile 2t+1 -> full-wave store.
        const int h   = h0 + 16 * half + n;
        const float v = (half ? c1[0] : c0[0]) + Wb[D_DIM * H_DIM + h]; // + bias row
        out[(size_t)b * H_DIM + h] = tanhf(v);
    }
}

extern "C" void kernel_launch(void* const* d_in, const int* in_sizes, int n_in,
                              void* d_out, int out_size, void* d_ws, size_t ws_size,
                              hipStream_t stream) {
    const float* X   = (const float*)d_in[0];
    const float* AUX = (const float*)d_in[1];
    float* out       = (float*)d_out;

    const int Btot = in_sizes[0] / D_DIM;               // 8192
    const int grid = (Btot + WAVES_PER_BLOCK - 1) / WAVES_PER_BLOCK;  // 1024

    DotLayer_wmma_kernel<<<grid, 32 * WAVES_PER_BLOCK, 0, stream>>>(X, AUX, out, Btot);
}